// DigitalGCN_12043088298065
// MI455X (gfx1250) — compile-verified
//
#include <hip/hip_runtime.h>
#include <cstdint>
#include <cstddef>

// MI455X / gfx1250, wave32. WMMA f32 16x16x4 path, pair-interleaved B tiles.

typedef float v2f __attribute__((ext_vector_type(2)));
typedef float v8f __attribute__((ext_vector_type(8)));

#define B_ 16
#define N_ 1024
#define D_ 128

// ---------------------------------------------------------------- small prep
__global__ __launch_bounds__(256) void wdiff_kernel(const float* __restrict__ wl,
                                                    const float* __restrict__ wr,
                                                    float* __restrict__ wd) {
  int i = blockIdx.x * 256 + threadIdx.x;
  if (i < D_ * D_) wd[i] = wl[i] - wr[i];
}

__global__ __launch_bounds__(32) void snm_kernel(const int* __restrict__ nm,
                                                 float* __restrict__ snm) {
  int b = blockIdx.x, lane = threadIdx.x;
  int s = 0;
  for (int j = lane; j < N_; j += 32) s += nm[b * N_ + j];
#pragma unroll
  for (int off = 16; off; off >>= 1) s += __shfl_xor(s, off, 32);
  if (lane == 0) snm[b] = (float)s;
}

// node_weight = sigmoid(node @ w_weight + b); u = mask * node_weight
__global__ __launch_bounds__(256) void weight_kernel(const float* __restrict__ node,
                                                     const int* __restrict__ nmask,
                                                     const float* __restrict__ ww,
                                                     const float* __restrict__ bw,
                                                     float* __restrict__ u,
                                                     float* __restrict__ nw) {
  int wave = threadIdx.x >> 5, lane = threadIdx.x & 31;
  int row0 = blockIdx.x * 64 + wave * 8;
  float w0 = ww[lane], w1 = ww[lane + 32], w2 = ww[lane + 64], w3 = ww[lane + 96];
  float bb = bw[0];
  for (int k = 0; k < 8; ++k) {
    int row = row0 + k;
    const float* np = node + (size_t)row * D_;
    float acc = np[lane] * w0 + np[lane + 32] * w1 + np[lane + 64] * w2 + np[lane + 96] * w3;
#pragma unroll
    for (int off = 16; off; off >>= 1) acc += __shfl_xor(acc, off, 32);
    if (lane == 0) {
      float sw = 1.0f / (1.0f + __expf(-(acc + bb)));
      int b = row >> 10, n = row & (N_ - 1);
      nw[b * 2 * N_ + n] = sw;              // all_node_weight[b, iter, n]
      u[row] = (float)nmask[row] * sw;
    }
  }
}

// T[b,:] = sum_j u[b,j] * HR[b,j,:]   (deterministic, no atomics)
__global__ __launch_bounds__(128) void tvec_kernel(const float* __restrict__ u,
                                                   const float* __restrict__ HR,
                                                   float* __restrict__ T) {
  int b = blockIdx.x, d = threadIdx.x;
  const float* hb = HR + (size_t)b * N_ * D_ + d;
  const float* ub = u + b * N_;
  float acc = 0.f;
#pragma unroll 4
  for (int j = 0; j < N_; ++j) acc += ub[j] * hb[(size_t)j * D_];
  T[b * D_ + d] = acc;
}

// Stage a 64(K) x 128(col) f32 tile into LDS with K-pair interleave:
// element (k, col) -> lB[(k>>1)*256 + col*2 + (k&1)]  (conflict-free b128 stores)
__device__ __forceinline__ void stage_b_tile(float* lB, const float* __restrict__ src,
                                             int k0, int tid) {
  for (int i = tid; i < 32 * 32; i += 128) {
    int p = i >> 5, c4 = i & 31;                       // k-pair, 4-col group
    float4 r0 = *(const float4*)(src + (size_t)(k0 + 2 * p) * D_ + c4 * 4);
    float4 r1 = *(const float4*)(src + (size_t)(k0 + 2 * p + 1) * D_ + c4 * 4);
    float4 o0 = make_float4(r0.x, r1.x, r0.y, r1.y);
    float4 o1 = make_float4(r0.z, r1.z, r0.w, r1.w);
    *(float4*)&lB[p * 256 + c4 * 8]     = o0;
    *(float4*)&lB[p * 256 + c4 * 8 + 4] = o1;
  }
}

// ------------------------------------------------- projection GEMM (WMMA f32)
// out[M,128] = A[M,128] @ W[128,128] (+bias) (*rowscale)
// WG = 4 waves, 64 rows x 128 cols; each wave: 16 rows, 8 column tiles.
__global__ __launch_bounds__(128) void proj_kernel(const float* __restrict__ A,
                                                   const float* __restrict__ W,
                                                   const float* __restrict__ bias,
                                                   const float* __restrict__ rowscale,
                                                   float* __restrict__ out) {
  __shared__ float lA[64 * 64];    // 16 KB, row-major
  __shared__ float lB[64 * 128];   // 32 KB, K-pair interleaved
  const int wave = threadIdx.x >> 5, lane = threadIdx.x & 31;
  const int row0 = blockIdx.x * 64;
  const int m = lane & 15;
  const int kh = (lane >> 4) << 1;           // 0 or 2: K-pair per lane half
  v8f acc[8] = {};
  for (int k0 = 0; k0 < D_; k0 += 64) {
    for (int i = threadIdx.x; i < 64 * 16; i += 128) {   // A tile 64x64 f32
      int r = i >> 4, c4 = i & 15;
      *(float4*)&lA[r * 64 + c4 * 4] =
          *(const float4*)(A + (size_t)(row0 + r) * D_ + k0 + c4 * 4);
    }
    stage_b_tile(lB, W, k0, threadIdx.x);
    __syncthreads();
#pragma unroll
    for (int kk = 0; kk < 64; kk += 4) {
      // A 16x4: lanes0-15 rows M, K = kk+{0,1}; lanes16-31: K = kk+{2,3}
      v2f af = *(const v2f*)&lA[(wave * 16 + m) * 64 + kk + kh];
      const int pb = ((kk + kh) >> 1) * 256 + m * 2;
#pragma unroll
      for (int c = 0; c < 8; ++c) {
        v2f bf = *(const v2f*)&lB[pb + c * 32];   // one aligned ds_load_b64
        acc[c] = __builtin_amdgcn_wmma_f32_16x16x4_f32(
            false, af, false, bf, (short)0, acc[c], false, false);
      }
    }
    __syncthreads();
  }
  // C/D 16x16: VGPR v -> row M = v + (lane>=16 ? 8 : 0), col = lane&15
  const int rbase = row0 + wave * 16 + ((lane >> 4) << 3);
#pragma unroll
  for (int c = 0; c < 8; ++c) {
    int coln = c * 16 + m;
    float badd = bias ? bias[coln] : 0.f;
#pragma unroll
    for (int v = 0; v < 8; ++v) {
      int r = rbase + v;
      float s = acc[c][v] + badd;
      if (rowscale) s *= rowscale[r];
      out[(size_t)r * D_ + coln] = s;
    }
  }
}

// --------------------------------- adjacency GEMM + fused epilogue (per batch)
// S = G @ Hd' ; node_out = relu(Hself + fac*(S - g_ii*Hd'_i + T - u_i*HR_i))
__global__ __launch_bounds__(128) void agg_kernel(const int* __restrict__ graph,
                                                  const float* __restrict__ Hd,
                                                  const float* __restrict__ Hself,
                                                  const float* __restrict__ HR,
                                                  const float* __restrict__ u,
                                                  const float* __restrict__ T,
                                                  const int* __restrict__ nmask,
                                                  const float* __restrict__ snm,
                                                  float* __restrict__ out) {
  __shared__ int   lA[64 * 64];    // graph tile as int (cvt during frag load)
  __shared__ float lB[64 * 128];   // Hd' tile, K-pair interleaved
  const int b = blockIdx.y;
  const int row0 = blockIdx.x * 64;
  const int wave = threadIdx.x >> 5, lane = threadIdx.x & 31;
  const int m = lane & 15;
  const int kh = (lane >> 4) << 1;
  const int* gA = graph + ((size_t)b * N_ + row0) * N_;
  const float* Bb = Hd + (size_t)b * N_ * D_;
  v8f acc[8] = {};
  for (int k0 = 0; k0 < N_; k0 += 64) {
    for (int i = threadIdx.x; i < 64 * 16; i += 128) {   // graph tile 64x64 i32
      int r = i >> 4, c4 = i & 15;
      *(int4*)&lA[r * 64 + c4 * 4] =
          *(const int4*)(gA + (size_t)r * N_ + k0 + c4 * 4);
    }
    stage_b_tile(lB, Bb, k0, threadIdx.x);
    __syncthreads();
    if (k0 + 64 < N_)   // gfx1250 global_prefetch for next graph tile
      __builtin_prefetch(gA + (size_t)(threadIdx.x >> 1) * N_ + k0 + 64, 0, 0);
#pragma unroll
    for (int kk = 0; kk < 64; kk += 4) {
      int2 ai = *(const int2*)&lA[(wave * 16 + m) * 64 + kk + kh];  // ds_load_b64
      v2f af;
      af.x = (float)ai.x;   // cvt co-executes with WMMA
      af.y = (float)ai.y;
      const int pb = ((kk + kh) >> 1) * 256 + m * 2;
#pragma unroll
      for (int c = 0; c < 8; ++c) {
        v2f bf = *(const v2f*)&lB[pb + c * 32];
        acc[c] = __builtin_amdgcn_wmma_f32_16x16x4_f32(
            false, af, false, bf, (short)0, acc[c], false, false);
      }
    }
    __syncthreads();
  }
  const int rbase = row0 + wave * 16 + ((lane >> 4) << 3);
  float fac[8], uiv[8], giiv[8];
#pragma unroll
  for (int v = 0; v < 8; ++v) {
    int r = rbase + v;
    float nm = (float)nmask[b * N_ + r];
    float nn = nm * (snm[b] - nm);
    fac[v] = nm / (nn >= 1.f ? nn : 1.f);
    uiv[v] = u[b * N_ + r];
    giiv[v] = (float)graph[((size_t)b * N_ + r) * N_ + r];
  }
#pragma unroll
  for (int c = 0; c < 8; ++c) {
    int coln = c * 16 + m;
    float tb = T[b * D_ + coln];
#pragma unroll
    for (int v = 0; v < 8; ++v) {
      int r = rbase + v;
      size_t idx = ((size_t)b * N_ + r) * D_ + coln;
      float corr = acc[c][v] - giiv[v] * Hd[idx] + tb - uiv[v] * HR[idx];
      float val = Hself[idx] + fac[v] * corr;
      out[idx] = val > 0.f ? val : 0.f;
    }
  }
}

// --------------------------------------------------------------------- launch
extern "C" void kernel_launch(void* const* d_in, const int* in_sizes, int n_in,
                              void* d_out, int out_size, void* d_ws, size_t ws_size,
                              hipStream_t stream) {
  const float* node0  = (const float*)d_in[0];
  const int*   nmask  = (const int*)d_in[1];
  const int*   graph  = (const int*)d_in[2];
  const float* ww     = (const float*)d_in[3];
  const float* bw     = (const float*)d_in[4];
  const float* wself  = (const float*)d_in[5];
  const float* bself  = (const float*)d_in[6];
  const float* wleft  = (const float*)d_in[7];
  const float* wright = (const float*)d_in[8];
  float* out = (float*)d_out;
  float* nw_base = out + (size_t)B_ * N_ * D_;   // all_node_weight [B,2,N]

  float* ws = (float*)d_ws;
  size_t o = 0;
  float* wdiff = ws + o; o += (size_t)D_ * D_;
  float* snm   = ws + o; o += B_;
  float* u     = ws + o; o += (size_t)B_ * N_;
  float* T     = ws + o; o += (size_t)B_ * D_;
  float* Hself = ws + o; o += (size_t)B_ * N_ * D_;
  float* HR    = ws + o; o += (size_t)B_ * N_ * D_;
  float* Hd    = ws + o; o += (size_t)B_ * N_ * D_;
  float* nodeA = ws + o; o += (size_t)B_ * N_ * D_;   // ~33.7 MB total

  wdiff_kernel<<<dim3((D_ * D_ + 255) / 256), dim3(256), 0, stream>>>(wleft, wright, wdiff);
  snm_kernel<<<dim3(B_), dim3(32), 0, stream>>>(nmask, snm);

  const float* nin = node0;
  for (int it = 0; it < 2; ++it) {
    float* nout = (it == 0) ? nodeA : out;
    weight_kernel<<<dim3(B_ * N_ / 64), dim3(256), 0, stream>>>(nin, nmask, ww, bw, u,
                                                                nw_base + it * N_);
    proj_kernel<<<dim3(B_ * N_ / 64), dim3(128), 0, stream>>>(nin, wself, bself,
                                                              (const float*)nullptr, Hself);
    proj_kernel<<<dim3(B_ * N_ / 64), dim3(128), 0, stream>>>(nin, wright,
                                                              (const float*)nullptr,
                                                              (const float*)nullptr, HR);
    proj_kernel<<<dim3(B_ * N_ / 64), dim3(128), 0, stream>>>(nin, wdiff,
                                                              (const float*)nullptr, u, Hd);
    tvec_kernel<<<dim3(B_), dim3(128), 0, stream>>>(u, HR, T);
    agg_kernel<<<dim3(N_ / 64, B_), dim3(128), 0, stream>>>(graph, Hd, Hself, HR, u, T,
                                                            nmask, snm, nout);
    nin = nout;
  }
  (void)in_sizes; (void)n_in; (void)out_size; (void)ws_size;
}